// SparseOptFlowLK_75299366633637
// MI455X (gfx1250) — compile-verified
//
#include <hip/hip_runtime.h>
#include <math.h>

// Problem constants (match reference)
#define WINSZ 21
#define IMH   1008
#define IMW   1008
#define NHS   48
#define NWS   48
#define NT    (NHS * NWS)     // 2304 windows per image
#define NB    16
#define NPIX  (WINSZ * WINSZ) // 441

// exp(-r2/8) == exp2(-r2 * 0.125 * log2(e))
#define WEXP_C (-0.18033688011112042f)

typedef float v2f __attribute__((ext_vector_type(2)));
typedef float v8f __attribute__((ext_vector_type(8)));

__device__ __forceinline__ float fetch_pad(const float* __restrict__ img, int y, int x) {
    // SAME conv zero padding at image edges
    if ((unsigned)y < (unsigned)IMH && (unsigned)x < (unsigned)IMW)
        return img[y * IMW + x];
    return 0.0f;
}

// Bit-exact lane broadcast of a float (readlane is int-typed; avoid value conversion)
__device__ __forceinline__ float readlane_f32(float v, int srclane) {
    return __int_as_float(__builtin_amdgcn_readlane(__float_as_int(v), srclane));
}

template <bool BORDER>
__device__ __forceinline__ void process_pixel(const float* __restrict__ ff,
                                              const float* __restrict__ sf,
                                              int gy, int gx, int row, int col,
                                              float& a00, float& a01, float& a11,
                                              float& b0, float& b1)
{
    float fm1m1, fm10, fm1p1, f0m1, f00, f0p1, fp1m1, fp10, fp1p1, s00;
    if (BORDER) {
        fm1m1 = fetch_pad(ff, gy - 1, gx - 1);
        fm10  = fetch_pad(ff, gy - 1, gx    );
        fm1p1 = fetch_pad(ff, gy - 1, gx + 1);
        f0m1  = fetch_pad(ff, gy,     gx - 1);
        f00   = ff[gy * IMW + gx];
        f0p1  = fetch_pad(ff, gy,     gx + 1);
        fp1m1 = fetch_pad(ff, gy + 1, gx - 1);
        fp10  = fetch_pad(ff, gy + 1, gx    );
        fp1p1 = fetch_pad(ff, gy + 1, gx + 1);
        s00   = sf[gy * IMW + gx];
    } else {
        // Interior window: one base address, 9 immediate-offset b32 loads
        const int idx = gy * IMW + gx;
        const float* __restrict__ c = ff + idx;
        fm1m1 = c[-IMW - 1];
        fm10  = c[-IMW    ];
        fm1p1 = c[-IMW + 1];
        f0m1  = c[-1];
        f00   = c[0];
        f0p1  = c[1];
        fp1m1 = c[IMW - 1];
        fp10  = c[IMW    ];
        fp1p1 = c[IMW + 1];
        s00   = sf[idx];
    }

    // Scharr (cross-correlation, as in lax.conv)
    float Ix = 3.0f * (fm1p1 - fm1m1) + 10.0f * (f0p1 - f0m1) + 3.0f * (fp1p1 - fp1m1);
    float Iy = 3.0f * (fp1m1 - fm1m1) + 10.0f * (fp10 - fm10) + 3.0f * (fp1p1 - fm1p1);

    // Gaussian window weight, sigma = 2
    float dxc = (float)(col - 10);
    float dyc = (float)(row - 10);
    float r2  = dxc * dxc + dyc * dyc;
    float w   = exp2f(r2 * WEXP_C);   // single v_exp_f32

    float dt = (f00 - s00) * w;       // rhs = -(sf-ff)*w

    a00 += Ix * Ix * w;
    a01 += Ix * Iy * w;
    a11 += Iy * Iy * w;
    b0  += Ix * dt;
    b1  += Iy * dt;
}

__global__ __launch_bounds__(256) void lk_flow_kernel(const float* __restrict__ imgs,
                                                      float* __restrict__ out)
{
    const int lane = threadIdx.x & 31;
    const int wave = threadIdx.x >> 5;
    const int win  = blockIdx.x * 8 + wave;      // 0 .. NB*NT-1 (grid sized exactly)

    const int bi = win / NT;
    const int t  = win - bi * NT;
    const int ty = t / NWS;
    const int tx = t - ty * NWS;
    const int wy = ty * WINSZ;
    const int wx = tx * WINSZ;

    const float* __restrict__ ff = imgs + (size_t)(bi * 2 + 0) * (IMH * IMW);
    const float* __restrict__ sf = imgs + (size_t)(bi * 2 + 1) * (IMH * IMW);

    float a00 = 0.0f, a01 = 0.0f, a11 = 0.0f, b0 = 0.0f, b1 = 0.0f;

    // Incremental (row,col) tracking: pixel p = lane + 32*i, p = row*21 + col.
    int row = (lane >= WINSZ) ? 1 : 0;
    int col = (lane >= WINSZ) ? lane - WINSZ : lane;

    // Only windows touching the image edge need zero-padding (wave-uniform test).
    const bool border = (ty == 0) | (ty == NHS - 1) | (tx == 0) | (tx == NWS - 1);

    if (!border) {
        #pragma unroll
        for (int i = 0; i < 13; ++i) {           // p <= 415 < 441: all lanes valid
            process_pixel<false>(ff, sf, wy + row, wx + col, row, col,
                                 a00, a01, a11, b0, b1);
            col += 11; row += 1;                 // advance p by 32 = 21 + 11
            if (col >= WINSZ) { col -= WINSZ; row += 1; }
        }
        if (lane < 25)                           // tail: p = lane + 416 < 441
            process_pixel<false>(ff, sf, wy + row, wx + col, row, col,
                                 a00, a01, a11, b0, b1);
    } else {
        #pragma unroll
        for (int i = 0; i < 13; ++i) {
            process_pixel<true>(ff, sf, wy + row, wx + col, row, col,
                                a00, a01, a11, b0, b1);
            col += 11; row += 1;
            if (col >= WINSZ) { col -= WINSZ; row += 1; }
        }
        if (lane < 25)
            process_pixel<true>(ff, sf, wy + row, wx + col, row, col,
                                a00, a01, a11, b0, b1);
    }

    // --- Stage 1: xor-reduce 32 lanes -> 4 chunks per sum (chunk id = lane&3,
    //     value replicated across the 8 lanes sharing lane&3) ---
    #pragma unroll
    for (int m = 16; m >= 4; m >>= 1) {
        a00 += __shfl_xor(a00, m, 32);
        a01 += __shfl_xor(a01, m, 32);
        a11 += __shfl_xor(a11, m, 32);
        b0  += __shfl_xor(b0,  m, 32);
        b1  += __shfl_xor(b1,  m, 32);
    }

    // --- Stage 2: finish with V_WMMA_F32_16X16X4_F32 (ones x B column-sum).
    // Lane j (j>>2 = sum index, j&3 = chunk) exposes that sum's chunk.
    const int msel = lane >> 2;
    float chunk = (msel == 0) ? a00 :
                  (msel == 1) ? a01 :
                  (msel == 2) ? a11 :
                  (msel == 3) ? b0  :
                  (msel == 4) ? b1  : 0.0f;

    // B column n gets sum n's 4 chunks in its 4 physical slots
    // {v0/v1 x lanes n, n+16}; with A = ones, D[*][n] = total of sum n,
    // independent of the K-row-to-slot permutation.
    const int n    = lane & 15;
    const int src0 = (4 * n + 2 * (lane >> 4)) & 31;     // chunks 0/2
    v2f bmat;
    bmat.x = __shfl(chunk, src0, 32);
    bmat.y = __shfl(chunk, (src0 + 1) & 31, 32);         // chunks 1/3

    v2f amat;
    amat.x = 1.0f;
    amat.y = 1.0f;
    v8f cacc = {};
    cacc = __builtin_amdgcn_wmma_f32_16x16x4_f32(
        /*neg_a=*/false, amat, /*neg_b=*/false, bmat,
        /*c_mod=*/(short)0, cacc, /*reuse_a=*/false, /*reuse_b=*/false);

    // D VGPR0, lanes 0..15 = row M=0, cols N=0..15 -> lane n holds total n.
    float colsum = cacc[0];
    float A00 = readlane_f32(colsum, 0);
    float A01 = readlane_f32(colsum, 1);
    float A11 = readlane_f32(colsum, 2);
    float B0  = readlane_f32(colsum, 3);
    float B1  = readlane_f32(colsum, 4);

    // 2x2 solve (ATA symmetric)
    float det = A00 * A11 - A01 * A01;
    float inv = (det != 0.0f) ? (1.0f / det) : 0.0f;
    float Vx  = inv * ( A11 * B0 - A01 * B1);
    float Vy  = inv * (-A01 * B0 + A00 * B1);

    // window position + velocity remap
    float wpx = (float)(wx + 10) * (2.0f / (float)IMW) - 1.0f;
    float wpy = (float)(wy + 10) * (2.0f / (float)IMH) - 1.0f;
    float vx  = Vx * (1.0f / 48.0f) + wpx;   // CENTER_REL = [48, 48]
    float vy  = Vy * (1.0f / 48.0f) + wpy;

    if (lane == 0) {
        float4* o = (float4*)out;            // out[b][t] = {wpx, wpy, vx, vy}
        o[win] = make_float4(wpx, wpy, vx, vy);
    }
}

extern "C" void kernel_launch(void* const* d_in, const int* in_sizes, int n_in,
                              void* d_out, int out_size, void* d_ws, size_t ws_size,
                              hipStream_t stream) {
    const float* imgs = (const float*)d_in[0];
    float* out = (float*)d_out;

    // NB*NT = 36864 windows, 8 waves (one window each) per 256-thread block
    dim3 grid((NB * NT) / 8);
    dim3 block(256);
    hipLaunchKernelGGL(lk_flow_kernel, grid, block, 0, stream, imgs, out);
}